// Yolo_Loss_71038759076375
// MI455X (gfx1250) — compile-verified
//
#include <hip/hip_runtime.h>
#include <stdint.h>

#define GG   13
#define AA   5
#define NCLS 5
#define NGT  32
#define NCELL (GG*GG)            // 169
#define NBOX  (NCELL*AA)         // 845
#define PRED_PER_B (NBOX*(5+NCLS))  // 8450 floats per batch
#define PRED_BYTES (PRED_PER_B*4)   // 33800 bytes (8-byte granular)
#define THREADS 256

typedef __attribute__((ext_vector_type(2))) float v2f;
typedef __attribute__((ext_vector_type(8))) float v8f;

__constant__ float c_anc[AA][2] = {
    {2.5221f,  3.3145f },
    {3.19275f, 4.00944f},
    {4.5587f,  4.09892f},
    {5.47112f, 7.84053f},
    {6.2364f,  8.0071f }
};

__device__ __forceinline__ float sigm(float x) { return 1.0f / (1.0f + __expf(-x)); }

// Async DMA: global -> LDS, 8 bytes per enabled lane (ASYNCcnt-tracked).
__device__ __forceinline__ void async_g2l_b64(uint32_t lds_off, uint64_t gaddr) {
    asm volatile("global_load_async_to_lds_b64 %0, %1, off"
                 :: "v"(lds_off), "v"(gaddr)
                 : "memory");
}
__device__ __forceinline__ void wait_asynccnt0() {
    asm volatile("s_wait_asynccnt 0x0" ::: "memory");
}

// Barrier-free 256-way block reduction on the matrix pipe:
// 4 chained V_WMMA_F32_16X16X4_F32 with B == ones row-sum 64 partials each.
// Caller: all 256 threads must have stored into s_red and passed one
// __syncthreads() before wave 0 calls this. Result valid on all wave-0 lanes.
__device__ __forceinline__ float wave0_reduce256_wmma(const float* s_red, int lane) {
    const int row   = lane & 15;        // M index
    const int kbase = (lane >> 4) * 2;  // K = 0..1 (lanes 0-15) or 2..3 (lanes 16-31)
    v8f c = {};
    v2f bones; bones[0] = 1.0f; bones[1] = 1.0f;   // B = ones(4x16): layout-invariant
    #pragma unroll
    for (int chunk = 0; chunk < 4; ++chunk) {
        const float* src = &s_red[chunk * 64 + row * 4 + kbase];
        v2f a; a[0] = src[0]; a[1] = src[1];       // A[m][k] = s_red[chunk*64 + m*4 + k]
        c = __builtin_amdgcn_wmma_f32_16x16x4_f32(
                /*neg_a=*/false, a, /*neg_b=*/false, bones,
                /*c_mod=*/(short)0, c, /*reuse_a=*/false, /*reuse_b=*/false);
    }
    // Column n of D lives in lanes n and n+16 (8 VGPRs each); all columns equal.
    float s = c[0] + c[1] + c[2] + c[3] + c[4] + c[5] + c[6] + c[7];
    return s + __shfl_xor(s, 16, 32);
}

__global__ __launch_bounds__(THREADS)
void yolo_loss_stage1(const float* __restrict__ preds,
                      const float* __restrict__ gt_boxes,
                      const int*   __restrict__ gt_labels,
                      float* __restrict__ partial)
{
    __shared__ alignas(16) float s_pred[PRED_PER_B];  // 33800 B, filled by async DMA
    __shared__ float    s_resp[NBOX];                 // resp_mask per (cell,anchor)
    __shared__ float    s_respcell[NCELL];            // max over anchors
    __shared__ float    s_gxy[NCELL*2];               // gt_xy (broadcast over anchors)
    __shared__ float    s_gwh[NBOX*2];                // gt_wh per (cell,anchor)
    __shared__ uint32_t s_gcls[NBOX];                 // one-hot bitmask (NCLS<=5 bits)
    __shared__ float    s_gtc[NGT*4];                 // gt corners x1,y1,x2,y2
    __shared__ float    s_off[NGT*2];
    __shared__ float    s_bwbh[NGT*2];
    __shared__ int      s_cell[NGT];
    __shared__ int      s_j[NGT];
    __shared__ int      s_lab[NGT];
    __shared__ float    s_red[THREADS];

    const int b   = blockIdx.x;
    const int tid = threadIdx.x;

    // ---- (1) kick off async global->LDS copy of this batch's pred tile ----
    {
        const uint64_t gbase = (uint64_t)(uintptr_t)preds + (uint64_t)b * (uint64_t)PRED_BYTES;
        const uint32_t lbase = (uint32_t)(uintptr_t)(void*)&s_pred[0];
        for (int i = tid; i < PRED_BYTES / 8; i += THREADS) {
            async_g2l_b64(lbase + (uint32_t)(i * 8), gbase + (uint64_t)(i * 8));
        }
    }

    // ---- (2) zero target arrays (overlaps with DMA) ----
    for (int i = tid; i < NBOX; i += THREADS) {
        s_resp[i] = 0.0f;
        s_gcls[i] = 0u;
        s_gwh[2*i] = 0.0f; s_gwh[2*i+1] = 0.0f;
    }
    for (int i = tid; i < NCELL; i += THREADS) {
        s_respcell[i] = 0.0f;
        s_gxy[2*i] = 0.0f; s_gxy[2*i+1] = 0.0f;
    }

    // ---- (3) per-gt preprocessing: cell, offset, corners, best anchor ----
    if (tid < NGT) {
        const int n = tid;
        const size_t base = ((size_t)b * NGT + n) * 4;
        float bx = gt_boxes[base + 0] * (float)GG;
        float by = gt_boxes[base + 1] * (float)GG;
        float bw = gt_boxes[base + 2] * (float)GG;
        float bh = gt_boxes[base + 3] * (float)GG;
        float fx = floorf(bx), fy = floorf(by);
        int cx = min(max((int)fx, 0), GG - 1);
        int cy = min(max((int)fy, 0), GG - 1);
        s_cell[n] = cy * GG + cx;
        s_off[2*n]   = bx - fx;
        s_off[2*n+1] = by - fy;
        s_bwbh[2*n]   = bw;
        s_bwbh[2*n+1] = bh;
        float gx1 = bx - 0.5f*bw, gy1 = by - 0.5f*bh;
        float gx2 = bx + 0.5f*bw, gy2 = by + 0.5f*bh;
        s_gtc[4*n+0] = gx1; s_gtc[4*n+1] = gy1;
        s_gtc[4*n+2] = gx2; s_gtc[4*n+3] = gy2;
        float areaG = (gx2 - gx1) * (gy2 - gy1);
        float acx = fx + 0.5f, acy = fy + 0.5f;
        float best = -1.0f; int bj = 0;
        #pragma unroll
        for (int a = 0; a < AA; ++a) {
            float aw = c_anc[a][0], ah = c_anc[a][1];
            float ax1 = acx - 0.5f*aw, ay1 = acy - 0.5f*ah;
            float ax2 = acx + 0.5f*aw, ay2 = acy + 0.5f*ah;
            float iw = fmaxf(fminf(ax2, gx2) - fmaxf(ax1, gx1), 0.0f);
            float ih = fmaxf(fminf(ay2, gy2) - fmaxf(ay1, gy1), 0.0f);
            float inter = iw * ih;
            float areaA = (ax2 - ax1) * (ay2 - ay1);
            float iou = inter / fmaxf(areaA + areaG - inter, 1e-10f);
            if (iou > best) { best = iou; bj = a; }   // first-max tie-break
        }
        s_j[n]   = bj;
        s_lab[n] = gt_labels[(size_t)b * NGT + n];
    }
    __syncthreads();

    // ---- (4) sequential scatter (matches reference loop semantics exactly) ----
    if (tid == 0) {
        for (int n = 0; n < NGT; ++n) {
            int c = s_cell[n];
            int j = s_j[n];
            s_resp[c*AA + j] = 1.0f;
            s_respcell[c]    = 1.0f;
            s_gxy[2*c]   = s_off[2*n];         // set for all anchors of the cell
            s_gxy[2*c+1] = s_off[2*n+1];
            s_gwh[2*(c*AA + j)]   = s_bwbh[2*n]   / c_anc[j][0];
            s_gwh[2*(c*AA + j)+1] = s_bwbh[2*n+1] / c_anc[j][1];
            s_gcls[c*AA + j] |= (1u << s_lab[n]);  // one-hot bits accumulate
        }
    }

    // ---- (5) join DMA, then main sweep over the 845 pred boxes ----
    wait_asynccnt0();
    __syncthreads();

    float acc = 0.0f;
    for (int idx = tid; idx < NBOX; idx += THREADS) {
        const int a  = idx % AA;
        const int c  = idx / AA;
        const int gy = c / GG;   // first grid dim (row / y)
        const int gx = c % GG;   // second grid dim (col / x)
        const float* pp = &s_pred[idx * (5 + NCLS)];

        float px = sigm(pp[0]);
        float py = sigm(pp[1]);
        float pw = __expf(pp[2]);
        float ph = __expf(pp[3]);
        float pc = sigm(pp[4]);

        // softmax over 5 classes (max-subtracted, as jax.nn.softmax)
        float t0 = pp[5], t1 = pp[6], t2 = pp[7], t3 = pp[8], t4 = pp[9];
        float mx = fmaxf(fmaxf(fmaxf(t0, t1), fmaxf(t2, t3)), t4);
        float e0 = __expf(t0 - mx), e1 = __expf(t1 - mx), e2 = __expf(t2 - mx);
        float e3 = __expf(t3 - mx), e4 = __expf(t4 - mx);
        float inv = 1.0f / (e0 + e1 + e2 + e3 + e4);

        // predicted box corners
        float aw = c_anc[a][0], ah = c_anc[a][1];
        float w = aw * pw, h = ah * ph;
        float cxf = (float)gx + px, cyf = (float)gy + py;
        float x1 = cxf - 0.5f*w, y1 = cyf - 0.5f*h;
        float x2 = cxf + 0.5f*w, y2 = cyf + 0.5f*h;
        float areaP = (x2 - x1) * (y2 - y1);

        // gt_conf = max IoU against all 32 gt boxes (LDS broadcast reads)
        float gconf = 0.0f;
        #pragma unroll 4
        for (int n = 0; n < NGT; ++n) {
            float gx1 = s_gtc[4*n+0], gy1 = s_gtc[4*n+1];
            float gx2 = s_gtc[4*n+2], gy2 = s_gtc[4*n+3];
            float iw = fmaxf(fminf(x2, gx2) - fmaxf(x1, gx1), 0.0f);
            float ih = fmaxf(fminf(y2, gy2) - fmaxf(y1, gy1), 0.0f);
            float inter = iw * ih;
            float areaG = (gx2 - gx1) * (gy2 - gy1);
            float iou = inter / fmaxf(areaP + areaG - inter, 1e-10f);
            gconf = fmaxf(gconf, iou);
        }

        // loss terms
        float m    = s_resp[idx];
        float dx0  = s_gxy[2*c]   - px;
        float dx1  = s_gxy[2*c+1] - py;
        float dxy  = dx0*dx0 + dx1*dx1;
        float dw0  = sqrtf(s_gwh[2*idx])   - sqrtf(pw);
        float dw1  = sqrtf(s_gwh[2*idx+1]) - sqrtf(ph);
        float dwh  = dw0*dw0 + dw1*dw1;
        float dc   = gconf - pc;
        float dconf = dc * dc;
        uint32_t mask = s_gcls[idx];
        float p0 = e0*inv, p1 = e1*inv, p2 = e2*inv, p3 = e3*inv, p4 = e4*inv;
        float g0 = (float)((mask >> 0) & 1u), g1 = (float)((mask >> 1) & 1u);
        float g2 = (float)((mask >> 2) & 1u), g3 = (float)((mask >> 3) & 1u);
        float g4 = (float)((mask >> 4) & 1u);
        float dcls = (g0-p0)*(g0-p0) + (g1-p1)*(g1-p1) + (g2-p2)*(g2-p2)
                   + (g3-p3)*(g3-p3) + (g4-p4)*(g4-p4);

        acc += 5.0f * m * (dxy + dwh)
             + m * dconf
             + 0.5f * (1.0f - m) * dconf
             + s_respcell[c] * dcls;
    }

    // ---- (6) barrier-free block reduction on the matrix pipe ----
    s_red[tid] = acc;
    __syncthreads();
    if (tid < 32) {
        float tot = wave0_reduce256_wmma(s_red, tid);
        if (tid == 0) partial[b] = tot;
    }
}

__global__ __launch_bounds__(THREADS)
void yolo_loss_stage2(const float* __restrict__ partial, int n, float* __restrict__ out)
{
    __shared__ float s_red[THREADS];
    float acc = 0.0f;
    for (int i = threadIdx.x; i < n; i += THREADS) acc += partial[i];  // fixed order
    s_red[threadIdx.x] = acc;
    __syncthreads();
    if (threadIdx.x < 32) {
        float tot = wave0_reduce256_wmma(s_red, threadIdx.x);
        if (threadIdx.x == 0) out[0] = tot;
    }
}

extern "C" void kernel_launch(void* const* d_in, const int* in_sizes, int n_in,
                              void* d_out, int out_size, void* d_ws, size_t ws_size,
                              hipStream_t stream) {
    const float* preds     = (const float*)d_in[0];
    const float* gt_boxes  = (const float*)d_in[1];
    const int*   gt_labels = (const int*)d_in[2];
    float*       partial   = (float*)d_ws;

    const int B = in_sizes[0] / PRED_PER_B;   // 512 for the reference shapes

    yolo_loss_stage1<<<B, THREADS, 0, stream>>>(preds, gt_boxes, gt_labels, partial);
    yolo_loss_stage2<<<1, THREADS, 0, stream>>>(partial, B, (float*)d_out);
}